// LennardJonesCoulomb_20469814133141
// MI455X (gfx1250) — compile-verified
//
#include <hip/hip_runtime.h>

#define KCOUL 332.0636f
#define TILE_ELEMS 1024   // elements per tile (4 KB per array)

// first index i in [0,n) with a[i] >= key  (a sorted ascending)
__device__ __forceinline__ int lower_bound_i32(const int* __restrict__ a, int n, int key) {
    int lo = 0, hi = n;
    while (lo < hi) {
        int mid = (lo + hi) >> 1;
        if (a[mid] < key) lo = mid + 1; else hi = mid;
    }
    return lo;
}

// one b128 async copy: 16 B from (base + 4*gElem bytes) -> LDS byte offset ldsOff
#define ASYNC_B128(ldsOff, gElem, basePtr)                                       \
    asm volatile("global_load_async_to_lds_b128 %0, %1, %2"                      \
                 :: "v"(ldsOff), "v"((unsigned)(gElem) * 4u), "s"(basePtr)       \
                 : "memory")

#define WAIT_ASYNC(n)  asm volatile("s_wait_asynccnt " #n ::: "memory")
#define WAIT_DS0()     asm volatile("s_wait_dscnt 0x0" ::: "memory")

#define COMP4(v, j) ((j) == 0 ? (v).x : (j) == 1 ? (v).y : (j) == 2 ? (v).z : (v).w)

// One block per output frame. frames arrays are sorted, so each frame's
// contributions occupy a contiguous segment -> gather-reduce, no atomics, no
// streaming of the frames arrays (O(log P) scalar probes only). Segment data
// is streamed through LDS with double-buffered CDNA5 async-to-LDS copies,
// pipelined purely on ASYNCcnt/DScnt (thread-private LDS slots -> no barriers).
__global__ __launch_bounds__(256) void lj_coulomb_seg_kernel(
    const float* __restrict__ rc,  const float* __restrict__ q1c, const float* __restrict__ q2c,
    const int*   __restrict__ fc,
    const float* __restrict__ rl,  const float* __restrict__ e1,  const float* __restrict__ e2,
    const float* __restrict__ s1,  const float* __restrict__ s2,
    const int*   __restrict__ fl,
    float* __restrict__ out, int P)
{
    const int f  = (int)blockIdx.x;
    const int t  = (int)threadIdx.x;

    // 4 independent scalar binary-search chains (SMEM probes).
    const int cLo = lower_bound_i32(fc, P, f);
    const int cHi = lower_bound_i32(fc, P, f + 1);
    const int lLo = lower_bound_i32(fl, P, f);
    const int lHi = lower_bound_i32(fl, P, f + 1);

    // [buffer 0|1][array 0..4][TILE_ELEMS]; thread t owns float slots 4t..4t+3.
    __shared__ __align__(16) float lds[2 * 5 * TILE_ELEMS];
    const unsigned ldsBase = (unsigned)(unsigned long long)(&lds[0]);
    const unsigned myQuad  = 4u * (unsigned)t;
    const int      last    = P - 4;     // max legal b128 start element

    float acc = 0.0f;

    // ---------------- Coulomb: K * q1 * q2 / r over [cLo, cHi) ----------------
    {
        const int loA    = cLo & ~3;                 // 16B-aligned start
        const int len    = cHi - loA;
        const int ntiles = (len > 0) ? ((len + TILE_ELEMS - 1) >> 10) : 0;

        auto issue = [&](int k) {
            const int base = loA + (k << 10);
            int g = base + 4 * t;
            g = (g > last) ? last : g;               // keep load in-bounds
            WAIT_DS0();                              // prior reads of this buffer done
            const unsigned b = (unsigned)(k & 1);
            ASYNC_B128(ldsBase + ((b * 5u + 0u) * TILE_ELEMS + myQuad) * 4u, g, rc);
            ASYNC_B128(ldsBase + ((b * 5u + 1u) * TILE_ELEMS + myQuad) * 4u, g, q1c);
            ASYNC_B128(ldsBase + ((b * 5u + 2u) * TILE_ELEMS + myQuad) * 4u, g, q2c);
        };

        if (ntiles > 0) issue(0);
        for (int k = 0; k < ntiles; ++k) {
            if (k + 1 < ntiles) { issue(k + 1); WAIT_ASYNC(0x3); }
            else                {               WAIT_ASYNC(0x0); }
            const int b  = k & 1;
            const int i0 = loA + (k << 10) + 4 * t;  // unclamped quad start
            const float4 r4 = *(const float4*)(&lds[(b * 5 + 0) * TILE_ELEMS + 4 * t]);
            const float4 a4 = *(const float4*)(&lds[(b * 5 + 1) * TILE_ELEMS + 4 * t]);
            const float4 b4 = *(const float4*)(&lds[(b * 5 + 2) * TILE_ELEMS + 4 * t]);
            #pragma unroll
            for (int j = 0; j < 4; ++j) {
                const int i = i0 + j;
                if (i >= cLo && i < cHi)
                    acc += (KCOUL * COMP4(a4, j) * COMP4(b4, j)) / COMP4(r4, j);
            }
        }
    }

    // ---------- Lennard-Jones (Lorentz-Berthelot) over [lLo, lHi) ------------
    {
        const int loA    = lLo & ~3;
        const int len    = lHi - loA;
        const int ntiles = (len > 0) ? ((len + TILE_ELEMS - 1) >> 10) : 0;

        auto issue = [&](int k) {
            const int base = loA + (k << 10);
            int g = base + 4 * t;
            g = (g > last) ? last : g;
            WAIT_DS0();
            const unsigned b = (unsigned)(k & 1);
            ASYNC_B128(ldsBase + ((b * 5u + 0u) * TILE_ELEMS + myQuad) * 4u, g, rl);
            ASYNC_B128(ldsBase + ((b * 5u + 1u) * TILE_ELEMS + myQuad) * 4u, g, e1);
            ASYNC_B128(ldsBase + ((b * 5u + 2u) * TILE_ELEMS + myQuad) * 4u, g, e2);
            ASYNC_B128(ldsBase + ((b * 5u + 3u) * TILE_ELEMS + myQuad) * 4u, g, s1);
            ASYNC_B128(ldsBase + ((b * 5u + 4u) * TILE_ELEMS + myQuad) * 4u, g, s2);
        };

        if (ntiles > 0) issue(0);
        for (int k = 0; k < ntiles; ++k) {
            if (k + 1 < ntiles) { issue(k + 1); WAIT_ASYNC(0x5); }
            else                {               WAIT_ASYNC(0x0); }
            const int b  = k & 1;
            const int i0 = loA + (k << 10) + 4 * t;
            const float4 r4  = *(const float4*)(&lds[(b * 5 + 0) * TILE_ELEMS + 4 * t]);
            const float4 e14 = *(const float4*)(&lds[(b * 5 + 1) * TILE_ELEMS + 4 * t]);
            const float4 e24 = *(const float4*)(&lds[(b * 5 + 2) * TILE_ELEMS + 4 * t]);
            const float4 s14 = *(const float4*)(&lds[(b * 5 + 3) * TILE_ELEMS + 4 * t]);
            const float4 s24 = *(const float4*)(&lds[(b * 5 + 4) * TILE_ELEMS + 4 * t]);
            #pragma unroll
            for (int j = 0; j < 4; ++j) {
                const int i = i0 + j;
                if (i >= lLo && i < lHi) {
                    float eps = sqrtf(COMP4(e14, j) * COMP4(e24, j));
                    float sig = 0.5f * (COMP4(s14, j) + COMP4(s24, j));
                    float x   = sig / COMP4(r4, j);
                    float x2  = x * x;
                    float x6  = x2 * x2 * x2;
                    acc += 4.0f * eps * (x6 * x6 - x6);
                }
            }
        }
    }

    // --------------- block reduction, wave32 (warpSize == 32) ----------------
    for (int o = 16; o > 0; o >>= 1) acc += __shfl_down(acc, o, 32);
    __shared__ float wsum[8];
    const int wid = t >> 5, lid = t & 31;
    if (lid == 0) wsum[wid] = acc;
    __syncthreads();
    if (wid == 0) {
        float v = (lid < 8) ? wsum[lid] : 0.0f;
        for (int o = 16; o > 0; o >>= 1) v += __shfl_down(v, o, 32);
        if (lid == 0) out[f] = v;
    }
}

extern "C" void kernel_launch(void* const* d_in, const int* in_sizes, int n_in,
                              void* d_out, int out_size, void* d_ws, size_t ws_size,
                              hipStream_t stream) {
    const float* rc  = (const float*)d_in[0];
    const float* q1c = (const float*)d_in[1];
    const float* q2c = (const float*)d_in[2];
    const int*   fc  = (const int*)  d_in[3];
    const float* rl  = (const float*)d_in[4];
    const float* e1  = (const float*)d_in[5];
    const float* e2  = (const float*)d_in[6];
    const float* s1  = (const float*)d_in[7];
    const float* s2  = (const float*)d_in[8];
    const int*   fl  = (const int*)  d_in[9];

    const int P       = in_sizes[0];
    const int nframes = out_size;   // 4096; one block per frame

    dim3 grid((unsigned)nframes), block(256);
    lj_coulomb_seg_kernel<<<grid, block, 0, stream>>>(
        rc, q1c, q2c, fc, rl, e1, e2, s1, s2, fl, (float*)d_out, P);
}